// Simulation_76656576299659
// MI455X (gfx1250) — compile-verified
//
#include <hip/hip_runtime.h>
#include <stdint.h>

// ---------------------------------------------------------------------------
// Photophysics Markov-chain simulation, CDNA5 (gfx1250).
// One chain per lane; 1024 sequential events per chain.
// Streams u_choice/u_dwell through LDS with global_load_async_to_lds_b32
// (ASYNCcnt-tracked DMA, triple buffered -> 2 chunks / 32 DMAs in flight per
// wave). No barriers: each lane consumes only the LDS slot it requested.
// Read stream tagged TH_LOAD_NT, output stream TH_STORE_NT, keeping L2 free
// for the 800 KB bins table used by the searchsorted fixup probes.
// ---------------------------------------------------------------------------

#define EVENTS   1024
#define NSAMP    32768
#define BLOCK    256
#define CHUNK    8
#define NCHUNK   (EVENTS / CHUNK)
#define NBUF     3

// ---- CDNA5 async global->LDS DMA (CDNA5 ISA) --------------
__device__ __forceinline__ void async_ld_b32_nt(uint32_t lds_byte_off,
                                                uint64_t sbase,
                                                uint32_t gbyte_off) {
  // GVS mode: mem_addr = SADDR_u64 + VADDR_i32 ; LDS dest addr in first VGPR.
  // TH_LOAD_NT: single-use stream, do not retain in caches.
  asm volatile("global_load_async_to_lds_b32 %0, %1, %2 th:TH_LOAD_NT"
               :: "v"(lds_byte_off), "v"(gbyte_off), "s"(sbase)
               : "memory");
}

template <int N>
__device__ __forceinline__ void wait_asynccnt() {
  asm volatile("s_wait_asynccnt %0" :: "n"(N) : "memory");
}

__global__ __launch_bounds__(BLOCK)
void Simulation_76656576299659_kernel(const float* __restrict__ u_choice,
                                      const float* __restrict__ u_dwell,
                                      const float* __restrict__ probs_cumsum, // (4,5)
                                      const float* __restrict__ rates,        // (4,)
                                      const float* __restrict__ photon_flag,  // (5,)
                                      const float* __restrict__ bins,         // (nb,)
                                      const int*   __restrict__ trans_next,   // (5,)
                                      float* __restrict__ out,
                                      int nb) {
  __shared__ float sUC[NBUF][CHUNK * BLOCK];
  __shared__ float sUD[NBUF][CHUNK * BLOCK];

  const int lane  = threadIdx.x;
  const int chain = blockIdx.x * BLOCK + lane;

  // ---- hoist tiny LUTs into registers (uniform -> scalar loads) ----------
  float c[4][5];
#pragma unroll
  for (int st = 0; st < 4; ++st)
#pragma unroll
    for (int k = 0; k < 5; ++k) c[st][k] = probs_cumsum[st * 5 + k];
  float rt[4];
#pragma unroll
  for (int st = 0; st < 4; ++st) rt[st] = rates[st];
  float pf[5];
  int   tn[5];
#pragma unroll
  for (int k = 0; k < 5; ++k) { pf[k] = photon_flag[k]; tn[k] = trans_next[k]; }

  const float inv_dx = 1.0f / (bins[1] - bins[0]);   // uniform bin guess scale

  float* __restrict__ out_ph = out;
  float* __restrict__ out_ct = out + (size_t)EVENTS * NSAMP;
  float* __restrict__ out_v  = out + 2 * (size_t)EVENTS * NSAMP;

  const uint64_t base_uc = (uint64_t)(uintptr_t)u_choice;
  const uint64_t base_ud = (uint64_t)(uintptr_t)u_dwell;

  // Issue a chunk of CHUNK rows into buffer b (16 async DMA ops per wave).
  auto issue_chunk = [&](int ck, int b) {
#pragma unroll
    for (int r = 0; r < CHUNK; ++r) {
      const uint32_t goff = (uint32_t)(((ck * CHUNK + r) * NSAMP + chain) * 4u);
      const uint32_t l_uc = (uint32_t)(uintptr_t)&sUC[b][r * BLOCK + lane];
      const uint32_t l_ud = (uint32_t)(uintptr_t)&sUD[b][r * BLOCK + lane];
      async_ld_b32_nt(l_uc, base_uc, goff);
      async_ld_b32_nt(l_ud, base_ud, goff);
    }
  };

  int   s  = 0;     // current state (all chains start in S0)
  float ct = 0.0f;  // cumulative time

  // Prologue: fill 2 of the 3 buffers.
  issue_chunk(0, 0);
  issue_chunk(1, 1);

  int buf  = 0;  // buffer holding chunk ck
  int nbuf = 2;  // buffer the next issued chunk goes to

  for (int ck = 0; ck < NCHUNK; ++ck) {
    if (ck + 2 < NCHUNK) {
      issue_chunk(ck + 2, nbuf);
      wait_asynccnt<4 * CHUNK>();   // allow 2 chunks (32 DMAs) in flight
    } else if (ck + 1 < NCHUNK) {
      wait_asynccnt<2 * CHUNK>();   // 1 chunk still in flight
    } else {
      wait_asynccnt<0>();
    }

#pragma unroll
    for (int r = 0; r < CHUNK; ++r) {
      const int t = ck * CHUNK + r;
      const float uc = sUC[buf][r * BLOCK + lane];
      const float ud = sUD[buf][r * BLOCK + lane];

      // select CDF row + rate of current state (2-level cndmask tree)
      const bool sl2 = (s < 2);
      const bool is0 = (s == 0);
      const bool is2 = (s == 2);
      float cdf[5];
#pragma unroll
      for (int k = 0; k < 5; ++k)
        cdf[k] = sl2 ? (is0 ? c[0][k] : c[1][k]) : (is2 ? c[2][k] : c[3][k]);
      const float rate = sl2 ? (is0 ? rt[0] : rt[1]) : (is2 ? rt[2] : rt[3]);

      // inverse-CDF categorical: cat = #(uc >= cdf[k])
      int cat = 0;
#pragma unroll
      for (int k = 0; k < 5; ++k) cat += (uc >= cdf[k]) ? 1 : 0;
      if (cat > 4) cat = 4;

      // exponential dwell time
      const float dwell = -log1pf(-ud) / rate;
      ct += dwell;

      // photon flag + next state, selected by category
      const float ph = (cat == 0) ? pf[0]
                     : (cat == 1) ? pf[1]
                     : (cat == 2) ? pf[2]
                     : (cat == 3) ? pf[3] : pf[4];
      s = (cat == 0) ? tn[0]
        : (cat == 1) ? tn[1]
        : (cat == 2) ? tn[2]
        : (cat == 3) ? tn[3] : tn[4];

      // searchsorted(bins, ct, side='left'): arithmetic guess + exact fixup
      const float x = ct;
      int k = (int)fminf(fmaxf(ceilf(x * inv_dx), 0.0f), (float)nb);
      while (k > 0 && bins[k - 1] >= x) --k;
      while (k < nb && bins[k] < x) ++k;

      const size_t idx = (size_t)t * (size_t)NSAMP + (size_t)chain;
      __builtin_nontemporal_store(ph, out_ph + idx);
      __builtin_nontemporal_store(x, out_ct + idx);
      __builtin_nontemporal_store((float)k * ph, out_v + idx);
    }

    // rotate ring buffers (no modulo in hot loop)
    buf  = (buf  == NBUF - 1) ? 0 : buf + 1;
    nbuf = (nbuf == NBUF - 1) ? 0 : nbuf + 1;
  }
}

extern "C" void kernel_launch(void* const* d_in, const int* in_sizes, int n_in,
                              void* d_out, int out_size, void* d_ws, size_t ws_size,
                              hipStream_t stream) {
  const float* u_choice     = (const float*)d_in[0];
  const float* u_dwell      = (const float*)d_in[1];
  const float* probs_cumsum = (const float*)d_in[2];
  const float* rates        = (const float*)d_in[3];
  const float* photon_flag  = (const float*)d_in[4];
  const float* bins         = (const float*)d_in[5];
  const int*   trans_next   = (const int*)d_in[6];
  const int    nb           = in_sizes[5];

  dim3 grid(NSAMP / BLOCK);
  dim3 block(BLOCK);
  Simulation_76656576299659_kernel<<<grid, block, 0, stream>>>(
      u_choice, u_dwell, probs_cumsum, rates, photon_flag, bins, trans_next,
      (float*)d_out, nb);
}